// Brain_connectomic_graph_86139864089145
// MI455X (gfx1250) — compile-verified
//
#include <hip/hip_runtime.h>
#include <stdint.h>

typedef __attribute__((ext_vector_type(2))) float v2f;
typedef __attribute__((ext_vector_type(8))) float v8f;

#define NPG    100     // nodes per graph
#define EPG    2000    // edges per graph
#define LDADJ  100     // row stride of adjacency / X buffers (floats)
#define LDH    64      // row stride of H buffers
#define AROWS  112     // padded rows for A-role buffers (7 * 16)
#define BROWS  104     // padded rows for B-role buffers
#define NWAVE  8       // 256 threads / wave32

// ---------------- WMMA helpers: V_WMMA_F32_16X16X4_F32 -------------------
// D(16x16 f32) = A(16x4 f32) * B(4x16 f32) + C
__device__ __forceinline__ v8f wmma4(v2f a, v2f b, v8f c) {
  // (neg_a, A, neg_b, B, c_mod, C, reuse_a, reuse_b)
  return __builtin_amdgcn_wmma_f32_16x16x4_f32(false, a, false, b, (short)0, c,
                                               false, false);
}

// A-matrix 16x4 layout (ISA 7.12.2): lanes 0-15 -> M=lane, K={0,1};
// lanes 16-31 -> M=lane-16, K={2,3}. Contiguous f32 pair per lane.
__device__ __forceinline__ v2f ldA(const float* A, int lda, int m, int k, int lane) {
  const float* p = A + (m + (lane & 15)) * lda + (k + ((lane >> 4) << 1));
  v2f r; r.x = p[0]; r.y = p[1]; return r;
}
// B-matrix 4x16: row K striped across lanes within a VGPR (mirror of A).
__device__ __forceinline__ v2f ldB_lds(const float* B, int ldb, int k, int n, int lane) {
  int kk = k + ((lane >> 4) << 1);
  int c  = n + (lane & 15);
  v2f r; r.x = B[kk * ldb + c]; r.y = B[(kk + 1) * ldb + c]; return r;
}
// Bounds-guarded B load straight from global weight matrices [K x N] row-major.
__device__ __forceinline__ v2f ldB_gl(const float* B, int ldb, int k, int n,
                                      int K, int N, int lane) {
  int kk = k + ((lane >> 4) << 1);
  int c  = n + (lane & 15);
  v2f r;
  r.x = (kk     < K && c < N) ? B[kk * ldb + c]       : 0.0f;
  r.y = (kk + 1 < K && c < N) ? B[(kk + 1) * ldb + c] : 0.0f;
  return r;
}

// -------- GEMM1: H = X @ W  with per-row-hemisphere weight selection -------
// A in LDS [>=m+16 rows, lda], W in global [K x Nreal]. C -> LDS, rows < BROWS.
// m-tiles 0..2 are pure-left rows (<48), 4..6 pure-right (>=64), tile 3 mixed.
__device__ __forceinline__ void gemm_sel(const float* A, int lda, int K,
                                         const float* Wl, const float* Wr,
                                         int Nreal, float* C, int wave, int lane) {
  int ntiles = (Nreal + 15) >> 4;
  int njobs  = 7 * ntiles;
  for (int job = wave; job < njobs; job += NWAVE) {
    int mt = job / ntiles, nt = job % ntiles;
    int m = mt << 4, n = nt << 4;
    v8f acc = {0.f, 0.f, 0.f, 0.f, 0.f, 0.f, 0.f, 0.f};
    int c  = n + (lane & 15);
    int rb = m + ((lane >> 4) << 3);
    if (mt != 3) {
      const float* W = (mt < 3) ? Wl : Wr;
      for (int k = 0; k < K; k += 4)
        acc = wmma4(ldA(A, lda, m, k, lane), ldB_gl(W, Nreal, k, n, K, Nreal, lane), acc);
#pragma unroll
      for (int i = 0; i < 8; ++i) {
        int r = rb + i;
        if (r < BROWS) C[r * LDH + c] = acc[i];
      }
    } else {                                    // mixed tile rows 48..63
      v8f acc2 = {0.f, 0.f, 0.f, 0.f, 0.f, 0.f, 0.f, 0.f};
      for (int k = 0; k < K; k += 4) {
        v2f a = ldA(A, lda, m, k, lane);
        acc  = wmma4(a, ldB_gl(Wl, Nreal, k, n, K, Nreal, lane), acc);
        acc2 = wmma4(a, ldB_gl(Wr, Nreal, k, n, K, Nreal, lane), acc2);
      }
#pragma unroll
      for (int i = 0; i < 8; ++i) {
        int r = rb + i;
        float v = (r < 50) ? acc[i] : acc2[i];
        if (r < BROWS) C[r * LDH + c] = v;
      }
    }
  }
}

// -------- GEMM2: out = Adj @ H + bias(hemisphere), leaky_relu --------------
// Adj in LDS [AROWS x LDADJ] (K = 100), H in LDS [BROWS x LDH]. C rows < AROWS.
__device__ __forceinline__ void gemm_agg(const float* Adj, const float* Bm,
                                         int Nreal, const float* bL, const float* bR,
                                         float* C, int wave, int lane) {
  int ntiles = (Nreal + 15) >> 4;
  int njobs  = 7 * ntiles;
  for (int job = wave; job < njobs; job += NWAVE) {
    int mt = job / ntiles, nt = job % ntiles;
    int m = mt << 4, n = nt << 4;
    v8f acc = {0.f, 0.f, 0.f, 0.f, 0.f, 0.f, 0.f, 0.f};
    for (int k = 0; k < NPG; k += 4)
      acc = wmma4(ldA(Adj, LDADJ, m, k, lane), ldB_lds(Bm, LDH, k, n, lane), acc);
    int c  = n + (lane & 15);
    int rb = m + ((lane >> 4) << 3);
#pragma unroll
    for (int i = 0; i < 8; ++i) {
      int r = rb + i;
      float v = acc[i];
      if (r < NPG && c < Nreal) v += (r < 50) ? bL[c] : bR[c];
      v = (v > 0.f) ? v : 0.01f * v;            // leaky_relu(0.01)
      C[r * LDH + c] = v;                       // r <= 111 < AROWS always
    }
  }
}

// ---------------------------------------------------------------------------
__global__ __launch_bounds__(256) void brain_gcn_kernel(
    const float* __restrict__ x,   const float* __restrict__ ea,
    const float* __restrict__ Wl1, const float* __restrict__ bl1,
    const float* __restrict__ Wr1, const float* __restrict__ br1,
    const float* __restrict__ Wl2, const float* __restrict__ bl2,
    const float* __restrict__ Wr2, const float* __restrict__ br2,
    const float* __restrict__ Wg,  const float* __restrict__ bg,
    const int* __restrict__ srcIdx, const int* __restrict__ dstIdx,
    float* __restrict__ out)
{
  // 146,240 bytes of LDS per workgroup (one graph per WGP)
  __shared__ __align__(16) float lds[11200 + 11200 + 7168 + 6656 + 336];
  float* Xs   = lds;                 // [112 x 100] X, later re-used as Af
  float* Ah   = lds + 11200;         // [112 x 100] block-diag hemisphere adjacency
  float* bufA = lds + 22400;         // [112 x 64]  h1 / h2 / h3 (A-role)
  float* bufB = lds + 29568;         // [104 x 64]  X@W selections (B-role)
  float* degL = lds + 36224;         // [112] deg -> dinv (left)
  float* degR = degL + 112;
  float* degF = degR + 112;
  float* Af   = Xs;                  // alias after X is consumed

  const int g    = blockIdx.x;
  const int tid  = threadIdx.x;
  const int lane = tid & 31;
  const int wave = tid >> 5;
  const int nodeBase = g * NPG;
  const int edgeBase = g * EPG;

  // P0: zero all LDS
  for (int i = tid; i < 36560; i += 256) lds[i] = 0.f;
  __syncthreads();

  // P1: stage X (100x100, contiguous 40KB block per graph) + self-loop degrees
  {
    const float4* xin = (const float4*)(x + (size_t)g * (NPG * 100));
    float4* xo = (float4*)Xs;
    for (int i = tid; i < 2500; i += 256) xo[i] = xin[i];
    if (tid < NPG) { degL[tid] = 1.f; degR[tid] = 1.f; degF[tid] = 1.f; }
  }
  __syncthreads();

  // P2: edge scan #1 -> hemisphere adjacency + all degrees (ds_add_f32)
  for (int e = tid; e < EPG; e += 256) {
    int idx = edgeBase + e;
    int s = srcIdx[idx] - nodeBase;
    int d = dstIdx[idx] - nodeBase;
    float w = ea[idx];
    atomicAdd(&degF[d], w);
    bool sl = s < 50, dl = d < 50;
    if (sl && dl)        { atomicAdd(&Ah[d * LDADJ + s], w); atomicAdd(&degL[d], w); }
    else if (!sl && !dl) { atomicAdd(&Ah[d * LDADJ + s], w); atomicAdd(&degR[d], w); }
  }
  __syncthreads();

  // P3: deg -> dinv (in place)
  if (tid < NPG) {
    float a = degL[tid]; degL[tid] = (a > 0.f) ? rsqrtf(a) : 0.f;
    float b = degR[tid]; degR[tid] = (b > 0.f) ? rsqrtf(b) : 0.f;
    float c = degF[tid]; degF[tid] = (c > 0.f) ? rsqrtf(c) : 0.f;
  }
  __syncthreads();

  // P4: normalize Ah: A[d][s] = dinv[d]*(raw + I)*dinv[s] within each hemisphere
  for (int i = tid; i < NPG * NPG; i += 256) {
    int r = i / NPG, c = i % NPG;
    int rr = r * LDADJ + c;
    bool rl = r < 50, cl = c < 50;
    if (rl == cl) {
      const float* dv = rl ? degL : degR;
      float raw = Ah[rr] + ((r == c) ? 1.f : 0.f);
      Ah[rr] = dv[r] * raw * dv[c];
    }
  }
  __syncthreads();

  // P5: layer1 GEMM1:  bufB = X @ (W_l1 | W_r1 per source row)
  gemm_sel(Xs, LDADJ, 100, Wl1, Wr1, 64, bufB, wave, lane);
  __syncthreads();

  // P6: layer1 GEMM2: bufA = lrelu(Ah @ bufB + b1); also free X -> zero Af
  gemm_agg(Ah, bufB, 64, bl1, br1, bufA, wave, lane);
  for (int i = tid; i < 11200; i += 256) Af[i] = 0.f;
  __syncthreads();

  // P7: edge scan #2 -> full adjacency into Af
  for (int e = tid; e < EPG; e += 256) {
    int idx = edgeBase + e;
    int s = srcIdx[idx] - nodeBase;
    int d = dstIdx[idx] - nodeBase;
    atomicAdd(&Af[d * LDADJ + s], ea[idx]);
  }
  __syncthreads();

  // P8: layer2 GEMM1 (bufB = h1 @ Wl2/Wr2) + normalize Af with dinv_f
  gemm_sel(bufA, LDH, 64, Wl2, Wr2, 20, bufB, wave, lane);
  for (int i = tid; i < NPG * NPG; i += 256) {
    int r = i / NPG, c = i % NPG;
    int rr = r * LDADJ + c;
    float raw = Af[rr] + ((r == c) ? 1.f : 0.f);
    Af[rr] = degF[r] * raw * degF[c];
  }
  __syncthreads();

  // P9: layer2 GEMM2: bufA = h2 = lrelu(Ah @ bufB + b2)
  gemm_agg(Ah, bufB, 20, bl2, br2, bufA, wave, lane);
  __syncthreads();

  // P10: layer3 GEMM1: bufB = h2 @ Wg (same weight both hemispheres)
  gemm_sel(bufA, LDH, 20, Wg, Wg, 20, bufB, wave, lane);
  __syncthreads();

  // P11: layer3 GEMM2: bufA = h3 = lrelu(Af @ bufB + bg)
  gemm_agg(Af, bufB, 20, bg, bg, bufA, wave, lane);
  __syncthreads();

  // P12: scatter-mean pool over the graph's 100 nodes -> out[g][0..19]
  if (tid < 20) {
    float s = 0.f;
    for (int r = 0; r < NPG; ++r) s += bufA[r * LDH + tid];
    out[(size_t)g * 20 + tid] = s / 100.0f;
  }
}

extern "C" void kernel_launch(void* const* d_in, const int* in_sizes, int n_in,
                              void* d_out, int out_size, void* d_ws, size_t ws_size,
                              hipStream_t stream) {
  (void)n_in; (void)d_ws; (void)ws_size; (void)out_size;
  const float* x   = (const float*)d_in[0];
  const float* ea  = (const float*)d_in[1];
  const float* Wl1 = (const float*)d_in[2];
  const float* bl1 = (const float*)d_in[3];
  const float* Wr1 = (const float*)d_in[4];
  const float* br1 = (const float*)d_in[5];
  const float* Wl2 = (const float*)d_in[6];
  const float* bl2 = (const float*)d_in[7];
  const float* Wr2 = (const float*)d_in[8];
  const float* br2 = (const float*)d_in[9];
  const float* Wg  = (const float*)d_in[10];
  const float* bg  = (const float*)d_in[11];
  const int*   ei  = (const int*)d_in[12];     // edge_index [2][E]
  const int E = in_sizes[1];                   // len(edge_attr) == E
  const int* src = ei;
  const int* dst = ei + E;
  const int G = in_sizes[0] / (100 * 100);     // N*100 features -> num graphs

  brain_gcn_kernel<<<G, 256, 0, stream>>>(x, ea, Wl1, bl1, Wr1, br1,
                                          Wl2, bl2, Wr2, br2, Wg, bg,
                                          src, dst, (float*)d_out);
}